// AxonalConnections_22608707846341
// MI455X (gfx1250) — compile-verified
//
#include <hip/hip_runtime.h>
#include <hip/hip_bf16.h>

// out[b,t] = sum_s adj[t,s] * (1.5*E[s]-0.5) * spikes[b,s]
// Dense fp32 WMMA GEMM: M=T=16384 (rows of adj), N=B=32 (batch), K=S=16384.
// HBM-bound on the 1.07 GB adjacency stream (~46 us floor @ 23.3 TB/s).

typedef float v2f __attribute__((ext_vector_type(2)));
typedef float v8f __attribute__((ext_vector_type(8)));

#define SDIM 16384   // source neurons (K)
#define TDIM 16384   // target neurons (M)
#define BDIM 32      // batch (N)

// Block = 128 threads = 4 waves. Each wave computes a 16(M) x 32(N) tile of D.
__global__ __launch_bounds__(128)
void axonal_wmma_f32(const float* __restrict__ spikes,   // [B, S]
                     const float* __restrict__ Emask,    // [S]
                     const float* __restrict__ adj,      // [T, S]
                     float* __restrict__ out)            // [B, T]
{
    const int lane = threadIdx.x & 31;
    const int wave = threadIdx.x >> 5;
    const int h    = lane >> 4;      // K-half selector (0: K0/K1, 1: K2/K3)
    const int l    = lane & 15;      // M row (A) / N col (B) within tile
    const int t0   = blockIdx.x * 64 + wave * 16;   // first target row of tile

    // A fragment source: adj[t0+l, k+2h .. k+2h+1]  (8B/lane, 8B aligned)
    const float* aP  = adj    + (size_t)(t0 + l) * SDIM + 2 * h;
    // B fragments: modT[kk, n] = coef[kk] * spikes[n, kk]
    const float* b0P = spikes + (size_t)(l)      * SDIM + 2 * h;   // n = l
    const float* b1P = spikes + (size_t)(l + 16) * SDIM + 2 * h;   // n = l+16
    const float* eP  = Emask + 2 * h;

    v8f acc0 = {0.f, 0.f, 0.f, 0.f, 0.f, 0.f, 0.f, 0.f};
    v8f acc1 = {0.f, 0.f, 0.f, 0.f, 0.f, 0.f, 0.f, 0.f};

    // K loop, 2 WMMA K-steps (K=4 each) per iteration.
    for (int k = 0; k < SDIM; k += 8) {
        // ---- K-step 0: columns k .. k+3 ----
        v2f a0 = *(const v2f*)(aP  + k);
        v2f s0 = *(const v2f*)(b0P + k);
        v2f s1 = *(const v2f*)(b1P + k);
        v2f e0 = *(const v2f*)(eP  + k);
        v2f c0 = e0 * 1.5f - 0.5f;           // excitatory +1, inhibitory -0.5
        v2f b00 = c0 * s0;
        v2f b01 = c0 * s1;
        acc0 = __builtin_amdgcn_wmma_f32_16x16x4_f32(
                   false, a0, false, b00, (short)0, acc0, false, false);
        acc1 = __builtin_amdgcn_wmma_f32_16x16x4_f32(
                   false, a0, false, b01, (short)0, acc1, false, false);

        // ---- K-step 1: columns k+4 .. k+7 ----
        v2f a1 = *(const v2f*)(aP  + k + 4);
        v2f s2 = *(const v2f*)(b0P + k + 4);
        v2f s3 = *(const v2f*)(b1P + k + 4);
        v2f e1 = *(const v2f*)(eP  + k + 4);
        v2f c1 = e1 * 1.5f - 0.5f;
        v2f b10 = c1 * s2;
        v2f b11 = c1 * s3;
        acc0 = __builtin_amdgcn_wmma_f32_16x16x4_f32(
                   false, a1, false, b10, (short)0, acc0, false, false);
        acc1 = __builtin_amdgcn_wmma_f32_16x16x4_f32(
                   false, a1, false, b11, (short)0, acc1, false, false);
    }

    // D layout: VGPR v holds D[M = v + 8h, N = l]. 8 consecutive M values per
    // lane are 8 consecutive t values -> one 32B (b128x2) contiguous store.
    float* o0 = out + (size_t)(l)      * TDIM + t0 + 8 * h;   // batch col l
    float* o1 = out + (size_t)(l + 16) * TDIM + t0 + 8 * h;   // batch col l+16
    *(v8f*)o0 = acc0;
    *(v8f*)o1 = acc1;
}

extern "C" void kernel_launch(void* const* d_in, const int* in_sizes, int n_in,
                              void* d_out, int out_size, void* d_ws, size_t ws_size,
                              hipStream_t stream) {
    (void)in_sizes; (void)n_in; (void)out_size; (void)d_ws; (void)ws_size;
    const float* spikes = (const float*)d_in[0];   // [32, 128, 128]
    const float* Emask  = (const float*)d_in[1];   // [128, 128]
    const float* adj    = (const float*)d_in[2];   // [16384, 16384]
    float* out = (float*)d_out;                    // [32, 128, 128]

    dim3 block(128, 1, 1);              // 4 waves -> 64 target rows per block
    dim3 grid(TDIM / 64, 1, 1);         // 256 blocks
    axonal_wmma_f32<<<grid, block, 0, stream>>>(spikes, Emask, adj, out);
}